// InstanSeg_54125177864461
// MI455X (gfx1250) — compile-verified
//
#include <hip/hip_runtime.h>
#include <math.h>

typedef float v2f __attribute__((ext_vector_type(2)));
typedef float v8f __attribute__((ext_vector_type(8)));

#define WS        128
#define IMG_W     1024
#define HW        (1024*1024)
#define E_CH      6
#define NPIX_PER_C (WS*WS)

__device__ __forceinline__ float sigmoidf(float z) {
    return 1.0f / (1.0f + __expf(-z));
}

// cross half-wave identity exchange: lane i <-> lane i^16, pure VALU (v_permlanex16_b32)
__device__ __forceinline__ float xhalf(float v) {
    int i = __builtin_bit_cast(int, v);
    int r = __builtin_amdgcn_permlanex16(i, i, 0x76543210u, 0xfedcba98u, false, false);
    return __builtin_bit_cast(float, r);
}

// One wave = one tile of 16 consecutive pixels (N axis). Features on M axis:
//   D = W^T (16 x K, A operand, loop-invariant) x feats (K x 16 pixels, B operand)
// so no per-tile transpose is ever needed; layer-to-layer handoff is 8 permlanex16.
__global__ __launch_bounds__(256) void instanseg_mlp_wmma(
    const float* __restrict__ X,      // (6,1024,1024)
    const float* __restrict__ SIG,    // (1,1024,1024)
    const float* __restrict__ Cc,     // (512,6)
    const float* __restrict__ W1,     // (7,16) row-major
    const float* __restrict__ B1,     // (16)
    const float* __restrict__ W2,     // (16,16)
    const float* __restrict__ B2,     // (16)
    const float* __restrict__ W3,     // (16,1)
    const float* __restrict__ B3,     // (1)
    const int*   __restrict__ CENT,   // (512,2)
    float* __restrict__ PROB,         // (512*128*128)
    int numTiles)
{
    const int  lane   = threadIdx.x & 31;
    const int  waveId = blockIdx.x * (blockDim.x >> 5) + (threadIdx.x >> 5);
    const int  nWaves = gridDim.x * (blockDim.x >> 5);
    const bool hi     = lane >= 16;
    const int  n      = lane & 15;      // lo lanes: out-feat M=n / pixel n ; hi: M=n (K+2 rows)
    const int  mrow   = hi ? 8 : 0;

    // ---- A-layout weights (M=out-feature across lanes, K across VGPR pair), loaded once ----
    // A 16x4 tile: lanes 0-15 hold K,K+1 ; lanes 16-31 hold K+2,K+3 ; M = lane&15
    v2f a1t0, a1t1, a2t[4];
    const int k0 = hi ? 2 : 0;
    a1t0.x = W1[(0 + k0) * 16 + n];          // W1^T[m][k] = W1[k*16+m]
    a1t0.y = W1[(1 + k0) * 16 + n];
    if (!hi) { a1t1.x = W1[4 * 16 + n]; a1t1.y = W1[5 * 16 + n]; }
    else     { a1t1.x = W1[6 * 16 + n]; a1t1.y = 0.0f; }          // K=7 zero pad
    #pragma unroll
    for (int t = 0; t < 4; ++t) {
        a2t[t].x = W2[(4 * t + k0) * 16 + n];
        a2t[t].y = W2[(4 * t + k0 + 1) * 16 + n];
    }
    // biases along M (per-VGPR constants), W3 along K (per-lane-half constants)
    float bias1[8], bias2[8], w3r[8];
    #pragma unroll
    for (int r = 0; r < 8; ++r) {
        bias1[r] = B1[r + mrow];
        bias2[r] = B2[r + mrow];
        w3r[r]   = W3[r + mrow];
    }
    const float b3v = B3[0];

    for (int tile = waveId; tile < numTiles; tile += nWaves) {
        const int pixBase = tile * 16;
        const int ci = pixBase >> 14;                    // / (128*128); tile never crosses centroid
        const int q  = (pixBase & (NPIX_PER_C - 1)) + n; // this lane's pixel (N = lane&15)
        const int wy = q >> 7, wx = q & (WS - 1);

        int cy = CENT[2 * ci], cx = CENT[2 * ci + 1];
        cy = min(max(cy, WS / 2), IMG_W - WS / 2);
        cx = min(max(cx, WS / 2), IMG_W - WS / 2);
        const int off = (cy - WS / 2 + wy) * IMG_W + (cx - WS / 2 + wx);

        // ---- layer 1 B operand: feats (K=channel rows, N=pixel), 2 tiles of K=4 ----
        // lo lanes carry rows {0,1}/{4,5}; hi lanes rows {2,3}/{6,pad}
        v2f b0, b1t;
        b0.x = X[(size_t)(k0)     * HW + off] - Cc[ci * E_CH + k0];
        b0.y = X[(size_t)(k0 + 1) * HW + off] - Cc[ci * E_CH + k0 + 1];
        if (!hi) { b1t.x = X[(size_t)4 * HW + off] - Cc[ci * E_CH + 4];
                   b1t.y = X[(size_t)5 * HW + off] - Cc[ci * E_CH + 5]; }
        else     { b1t.x = SIG[off];           // channel 6 = sigma (no centroid offset)
                   b1t.y = 0.0f; }             // channel 7 pad

        v8f acc;
        #pragma unroll
        for (int r = 0; r < 8; ++r) acc[r] = bias1[r];
        acc = __builtin_amdgcn_wmma_f32_16x16x4_f32(false, a1t0, false, b0,
                                                    (short)0, acc, false, false);
        acc = __builtin_amdgcn_wmma_f32_16x16x4_f32(false, a1t1, false, b1t,
                                                    (short)0, acc, false, false);

        // relu, then re-shape h1 (D-layout) into layer-2 B tiles via cross-half swaps
        float h[8], sh[8];
        #pragma unroll
        for (int r = 0; r < 8; ++r) h[r] = fmaxf(acc[r], 0.0f);
        #pragma unroll
        for (int r = 0; r < 8; ++r) sh[r] = xhalf(h[r]);

        v2f bt[4];
        bt[0].x = hi ? sh[2] : h[0];  bt[0].y = hi ? sh[3] : h[1];   // K rows 0..3
        bt[1].x = hi ? sh[6] : h[4];  bt[1].y = hi ? sh[7] : h[5];   // K rows 4..7
        bt[2].x = hi ? h[2]  : sh[0]; bt[2].y = hi ? h[3]  : sh[1];  // K rows 8..11
        bt[3].x = hi ? h[6]  : sh[4]; bt[3].y = hi ? h[7]  : sh[5];  // K rows 12..15

        // ---- layer 2: 4 WMMAs of K=4 ----
        v8f acc2;
        #pragma unroll
        for (int r = 0; r < 8; ++r) acc2[r] = bias2[r];
        #pragma unroll
        for (int t = 0; t < 4; ++t)
            acc2 = __builtin_amdgcn_wmma_f32_16x16x4_f32(false, a2t[t], false, bt[t],
                                                         (short)0, acc2, false, false);

        // ---- layer 3: per-lane 8-feat dot + one cross-half add ----
        // lane n (lo) holds feats 0-7 of pixel n; lane n+16 holds feats 8-15 of pixel n
        float s = 0.0f;
        #pragma unroll
        for (int r = 0; r < 8; ++r) s = fmaf(fmaxf(acc2[r], 0.0f), w3r[r], s);
        s += xhalf(s);                         // full 16-feat dot in both halves
        const float o = sigmoidf(s + b3v);
        // prob is a write-once stream: non-temporal store, coalesced 64B per wave
        if (!hi) __builtin_nontemporal_store(o, PROB + pixBase + n);
    }
}

// iidd = [instance id ; window rows ; window cols], pure index streams (int32)
__global__ __launch_bounds__(256) void instanseg_iidd(
    const int* __restrict__ CENT, int* __restrict__ OUT, int N)
{
    int p = blockIdx.x * blockDim.x + threadIdx.x;
    if (p >= N) return;
    const int ci = p >> 14;
    const int q  = p & (NPIX_PER_C - 1);
    const int wy = q >> 7, wx = q & (WS - 1);
    int cy = CENT[2 * ci], cx = CENT[2 * ci + 1];
    cy = min(max(cy, WS / 2), IMG_W - WS / 2);
    cx = min(max(cx, WS / 2), IMG_W - WS / 2);
    __builtin_nontemporal_store(ci,                OUT + p);
    __builtin_nontemporal_store(cy - WS / 2 + wy,  OUT + N + p);
    __builtin_nontemporal_store(cx - WS / 2 + wx,  OUT + 2 * N + p);
}

extern "C" void kernel_launch(void* const* d_in, const int* in_sizes, int n_in,
                              void* d_out, int out_size, void* d_ws, size_t ws_size,
                              hipStream_t stream) {
    const float* X    = (const float*)d_in[0];
    const float* SIG  = (const float*)d_in[1];
    const float* Cc   = (const float*)d_in[2];
    const float* W1   = (const float*)d_in[3];
    const float* B1   = (const float*)d_in[4];
    const float* W2   = (const float*)d_in[5];
    const float* B2   = (const float*)d_in[6];
    const float* W3   = (const float*)d_in[7];
    const float* B3   = (const float*)d_in[8];
    const int*   CENT = (const int*)d_in[9];

    const int C = in_sizes[9] / 2;            // 512
    const int N = C * NPIX_PER_C;             // 8,388,608 pixels
    const int numTiles = N / 16;

    float* PROB = (float*)d_out;
    int*   IIDD = (int*)d_out + N;            // 3*N int32 after the prob block

    instanseg_mlp_wmma<<<2048, 256, 0, stream>>>(X, SIG, Cc, W1, B1, W2, B2, W3, B3,
                                                 CENT, PROB, numTiles);
    instanseg_iidd<<<(N + 255) / 256, 256, 0, stream>>>(CENT, IIDD, N);
}